// VectorQuantizer_13692355739656
// MI455X (gfx1250) — compile-verified
//
#include <hip/hip_runtime.h>

#define NVEC    8192      // 8*32*32 latent vectors
#define VOCAB   8192
#define CDIM    32
#define BETA_C  0.25f
#define ALPHA_C 0.01f
#define EPS_C   1e-12f
#define FLT_BIG 3.402823466e38f
#define LOG2E_C 1.4426950408889634f
#define LN2_C   0.6931471805599453f

typedef float v2f __attribute__((ext_vector_type(2)));
typedef float v8f __attribute__((ext_vector_type(8)));

// ---------------------------------------------------------------------------
// Kernel 1: row-normalize z (with bchw->nc transpose gather) and emb.
// Also emit sum-of-squares of the *normalized* rows (reference recomputes
// ||zfn||^2 and ||en||^2 for the distance matrix).
// ---------------------------------------------------------------------------
__global__ void vq_normalize(const float* __restrict__ z, const float* __restrict__ emb,
                             float* __restrict__ zfn, float* __restrict__ en,
                             float* __restrict__ zsq, float* __restrict__ esq) {
  int t = blockIdx.x * blockDim.x + threadIdx.x;
  if (t < NVEC) {
    int b  = t >> 10;          // n = b*1024 + h*32 + w
    int hw = t & 1023;
    const float* src = z + (size_t)b * (CDIM * 1024) + hw;  // stride 1024 over c
    float v[CDIM];
    float s = 0.f;
#pragma unroll
    for (int c = 0; c < CDIM; ++c) { float x = src[(size_t)c * 1024]; v[c] = x; s += x * x; }
    float m = fmaxf(sqrtf(s), EPS_C);
    float s2 = 0.f;
#pragma unroll
    for (int c = 0; c < CDIM; ++c) { float q = v[c] / m; zfn[(size_t)t * CDIM + c] = q; s2 += q * q; }
    zsq[t] = s2;
  } else if (t < NVEC + VOCAB) {
    int vi = t - NVEC;
    const float* src = emb + (size_t)vi * CDIM;
    float v[CDIM];
    float s = 0.f;
#pragma unroll
    for (int c = 0; c < CDIM; ++c) { float x = src[c]; v[c] = x; s += x * x; }
    float m = fmaxf(sqrtf(s), EPS_C);
    float s2 = 0.f;
#pragma unroll
    for (int c = 0; c < CDIM; ++c) { float q = v[c] / m; en[(size_t)vi * CDIM + c] = q; s2 += q * q; }
    esq[vi] = s2;
  }
}

// ---------------------------------------------------------------------------
// Kernel 2: fused  dot = zfn . en^T  (f32 WMMA 16x16x4, K=32 = 8 chained
// WMMAs) + online argmin of distance + unshifted sum-exp of logits
// (logits are in [-1,1] because both operands are unit-normalized, so
// log(sum(exp(.))) needs no max shift: no overflow possible in fp32).
// Then per-row gather/normalize of emb[idx], losses and output write.
// One workgroup (8 waves) per 16-row tile of z; waves split the 512 column
// tiles round-robin.
// C/D layout: lane l, vgpr i -> row M = i + 8*(l>>4), col N = l&15.
// A (16x4 f32): lane l holds row (l&15), vgprs = K {2*(l>>4), 2*(l>>4)+1}.
// B (4x16 f32): lane l holds col (l&15), vgprs = K {2*(l>>4), 2*(l>>4)+1}.
// ---------------------------------------------------------------------------
__global__ void __launch_bounds__(256)
vq_main(const float* __restrict__ zfn, const float* __restrict__ en,
        const float* __restrict__ zsq, const float* __restrict__ esq,
        const float* __restrict__ emb,
        float* __restrict__ out, float* __restrict__ vq_row, float* __restrict__ cm_row) {
  __shared__ float sA[16 * CDIM];
  __shared__ float sZ[16];
  __shared__ float rMin[8][16];
  __shared__ int   rIdx[8][16];
  __shared__ float rSum[8][16];

  int r0 = blockIdx.x * 16;
  for (int i = threadIdx.x; i < 16 * CDIM; i += blockDim.x)
    sA[i] = zfn[(size_t)r0 * CDIM + i];
  if (threadIdx.x < 16) sZ[threadIdx.x] = zsq[r0 + threadIdx.x];
  __syncthreads();

  int wave = threadIdx.x >> 5;
  int lane = threadIdx.x & 31;
  int l16  = lane & 15;   // A row / B col / D col
  int lh   = lane >> 4;   // half select

  v2f a[8];
#pragma unroll
  for (int k = 0; k < 8; ++k) {
    const float* ap = &sA[l16 * CDIM + 4 * k + 2 * lh];
    a[k].x = ap[0]; a[k].y = ap[1];
  }
  float zr[8];
#pragma unroll
  for (int i = 0; i < 8; ++i) zr[i] = sZ[i + 8 * lh];

  float mind[8], sml[8];
  int   mi[8];
#pragma unroll
  for (int i = 0; i < 8; ++i) { mind[i] = FLT_BIG; mi[i] = 0; sml[i] = 0.f; }

  for (int t = wave; t < VOCAB / 16; t += 8) {
    int c0 = t * 16;
    const float* bp = en + (size_t)(c0 + l16) * CDIM + 2 * lh;
    v8f acc = {0.f, 0.f, 0.f, 0.f, 0.f, 0.f, 0.f, 0.f};
#pragma unroll
    for (int k = 0; k < 8; ++k) {
      v2f bv; bv.x = bp[4 * k]; bv.y = bp[4 * k + 1];
      acc = __builtin_amdgcn_wmma_f32_16x16x4_f32(
          /*neg_a=*/false, a[k], /*neg_b=*/false, bv,
          /*c_mod=*/(short)0, acc, /*reuse_a=*/false, /*reuse_b=*/false);
    }
    float ec  = esq[c0 + l16];
    int  gcol = c0 + l16;
#pragma unroll
    for (int i = 0; i < 8; ++i) {
      float dot = acc[i];
      float d   = (zr[i] + ec) - 2.0f * dot;
      if (d < mind[i]) { mind[i] = d; mi[i] = gcol; }   // strict <: first index wins
      float lg = ALPHA_C * dot + (1.0f - ALPHA_C) * dot; // in [-1, 1]
      sml[i] += exp2f(lg * LOG2E_C);                     // unshifted sum-exp (base 2)
    }
  }

  // Reduce across the 16 lanes of each half (xor masks stay within a half).
#pragma unroll
  for (int m = 1; m < 16; m <<= 1) {
#pragma unroll
    for (int i = 0; i < 8; ++i) {
      float od = __shfl_xor(mind[i], m, 32);
      int   oi = __shfl_xor(mi[i],   m, 32);
      float os = __shfl_xor(sml[i],  m, 32);
      if (od < mind[i] || (od == mind[i] && oi < mi[i])) { mind[i] = od; mi[i] = oi; }
      sml[i] += os;
    }
  }
  if (l16 == 0) {
#pragma unroll
    for (int i = 0; i < 8; ++i) {
      int r = i + 8 * lh;
      rMin[wave][r] = mind[i]; rIdx[wave][r] = mi[i]; rSum[wave][r] = sml[i];
    }
  }
  __syncthreads();

  // Per-row tail: combine 8 wave-partials, gather+normalize emb[idx],
  // losses, straight-through output with (n,c)->(b,c,h,w) transpose.
  if (threadIdx.x < 16) {
    int r = threadIdx.x;
    float bd = rMin[0][r]; int bi = rIdx[0][r];
    float S  = rSum[0][r];
    for (int w = 1; w < 8; ++w) {
      float od = rMin[w][r]; int oi = rIdx[w][r];
      if (od < bd || (od == bd && oi < bi)) { bd = od; bi = oi; }
      S += rSum[w][r];
    }
    int n = r0 + r;
    const float* er = emb + (size_t)bi * CDIM;   // raw emb gather
    float ev[CDIM];
    float s = 0.f;
#pragma unroll
    for (int c = 0; c < CDIM; ++c) { float x = er[c]; ev[c] = x; s += x * x; }
    float mnorm = fmaxf(sqrtf(s), EPS_C);

    const float* enr = en + (size_t)bi * CDIM;   // exact logit at idx
    float dot = 0.f;
#pragma unroll
    for (int c = 0; c < CDIM; ++c) dot += sA[r * CDIM + c] * enr[c];
    float lg  = ALPHA_C * dot + (1.0f - ALPHA_C) * dot;
    float lse = LN2_C * log2f(S);                // ln(sum(exp(logits)))
    cm_row[n] = lse - lg;                        // -log_softmax at idx

    float vq = 0.f;
    int bb = n >> 10, hw = n & 1023;
#pragma unroll
    for (int c = 0; c < CDIM; ++c) {
      float zq   = ev[c] / mnorm;
      float zn   = sA[r * CDIM + c];
      float diff = zq - zn;
      vq += diff * diff;
      out[((size_t)bb * CDIM + c) * 1024 + hw] = zn + diff;  // zn + sg(zq - zn)
    }
    vq_row[n] = vq;
  }
}

// ---------------------------------------------------------------------------
// Kernel 3: deterministic fixed-shape reduction of per-row loss partials.
// ---------------------------------------------------------------------------
__global__ void vq_finalize(const float* __restrict__ vq_row,
                            const float* __restrict__ cm_row,
                            float* __restrict__ out) {
  __shared__ float sv[256];
  __shared__ float sc[256];
  float a = 0.f, b = 0.f;
  for (int i = threadIdx.x; i < NVEC; i += 256) { a += vq_row[i]; b += cm_row[i]; }
  sv[threadIdx.x] = a; sc[threadIdx.x] = b;
  __syncthreads();
  for (int s = 128; s > 0; s >>= 1) {
    if (threadIdx.x < (unsigned)s) {
      sv[threadIdx.x] += sv[threadIdx.x + s];
      sc[threadIdx.x] += sc[threadIdx.x + s];
    }
    __syncthreads();
  }
  if (threadIdx.x == 0) {
    out[NVEC * CDIM + 0] = sv[0] / (float)(NVEC * CDIM);      // vq_loss
    out[NVEC * CDIM + 1] = BETA_C * (sc[0] / (float)NVEC);    // commit_loss
  }
}

extern "C" void kernel_launch(void* const* d_in, const int* in_sizes, int n_in,
                              void* d_out, int out_size, void* d_ws, size_t ws_size,
                              hipStream_t stream) {
  (void)in_sizes; (void)n_in; (void)out_size; (void)ws_size;
  const float* z   = (const float*)d_in[0];   // [8,32,32,32]
  const float* emb = (const float*)d_in[1];   // [8192,32]
  float* out = (float*)d_out;                 // 262144 + 2 floats
  float* ws  = (float*)d_ws;

  float* zfn    = ws;                   // 262144
  float* en     = zfn + NVEC * CDIM;    // 262144
  float* zsq    = en + VOCAB * CDIM;    // 8192
  float* esq    = zsq + NVEC;           // 8192
  float* vq_row = esq + VOCAB;          // 8192
  float* cm_row = vq_row + NVEC;        // 8192

  vq_normalize<<<dim3((NVEC + VOCAB) / 256), dim3(256), 0, stream>>>(z, emb, zfn, en, zsq, esq);
  vq_main<<<dim3(NVEC / 16), dim3(256), 0, stream>>>(zfn, en, zsq, esq, emb, out, vq_row, cm_row);
  vq_finalize<<<dim3(1), dim3(256), 0, stream>>>(vq_row, cm_row, out);
}